// ESN_146028888207
// MI455X (gfx1250) — compile-verified
//
#include <hip/hip_runtime.h>

#define T_STEPS 20000
#define R_DIM   2000
#define NB      63          // workgroups: 63 * 32 cols = 2016 >= 2000
#define BLOCK   256         // 8 waves (wave32)
#define LEAK    0.3f

// CDNA5 async copy: global -> LDS, tracked by ASYNCcnt (gfx1250 only).
// Per-lane: copies 16 bytes from global address saddr + g_byte_off to LDS address lds_byte_addr.
__device__ __forceinline__ void async_g2lds_b128(unsigned lds_byte_addr,
                                                 unsigned g_byte_off,
                                                 const void* saddr) {
  asm volatile("global_load_async_to_lds_b128 %0, %1, %2"
               :: "v"(lds_byte_addr), "v"(g_byte_off), "s"(saddr)
               : "memory");
}
__device__ __forceinline__ void wait_async0() {
  asm volatile("s_wait_asynccnt 0" ::: "memory");
}

__global__ __launch_bounds__(BLOCK) void esn_persistent(
    const float* __restrict__ x,      // [T,1]
    const float* __restrict__ Win,    // [2,R] row-major
    const float* __restrict__ W,      // [R,R] row-major (h @ W: col j contiguous)
    const float* __restrict__ gain,   // [R]
    const float* __restrict__ bias,   // [R]
    float*       __restrict__ states, // [T,R] output
    float*       __restrict__ hbuf,   // [2,R] double-buffered state (zeroed)
    unsigned*    __restrict__ bar)    // global barrier counter (zeroed)
{
  __shared__ float  hs[R_DIM];        // snapshot of h for this step (8000 B)
  __shared__ float4 part[BLOCK];      // partial sums (4096 B)

  const int tid = threadIdx.x;
  const int bid = blockIdx.x;
  const int cg  = tid & 7;            // column group: 8 groups * 4 cols = 32 cols/WG
  const int ks  = tid >> 3;           // k-slice: 32 slices * 63 k each
  const int c0  = bid * 32 + cg * 4;  // first of 4 contiguous columns
  const bool colvalid = (c0 < R_DIM); // 2000 % 4 == 0 -> all-or-nothing per float4

  // Writer lanes (ks==0 -> tids 0..7, all in wave 0) preload per-column params.
  float p_w0[4] = {0.f,0.f,0.f,0.f}, p_w1[4] = {0.f,0.f,0.f,0.f};
  float p_g [4] = {0.f,0.f,0.f,0.f}, p_b [4] = {0.f,0.f,0.f,0.f};
  if (ks == 0 && colvalid) {
#pragma unroll
    for (int i = 0; i < 4; ++i) {
      p_w0[i] = Win[c0 + i];            // x weight
      p_w1[i] = Win[R_DIM + c0 + i];    // bias column (xb = [x, 1])
      p_g [i] = gain[c0 + i];
      p_b [i] = bias[c0 + i];
    }
  }

  const unsigned lds_base = (unsigned)(size_t)(&hs[0]); // low 32 bits == LDS offset

  for (int t = 0; t < T_STEPS; ++t) {
    // ---- async broadcast of h (500 float4) into LDS ----
    const float* hsrc = hbuf + (size_t)(t & 1) * R_DIM;
    {
      int idx = tid;                       // always < 500
      async_g2lds_b128(lds_base + 16u * (unsigned)idx, 16u * (unsigned)idx, hsrc);
      int idx2 = tid + BLOCK;
      if (idx2 < R_DIM / 4)
        async_g2lds_b128(lds_base + 16u * (unsigned)idx2, 16u * (unsigned)idx2, hsrc);
      wait_async0();
    }
    __syncthreads();

    // ---- partial mat-vec: this thread covers k in [kbeg,kend), 4 columns ----
    float ax = 0.f, ay = 0.f, az = 0.f, aw = 0.f;
    if (colvalid) {
      int kbeg = ks * 63;
      int kend = kbeg + 63; if (kend > R_DIM) kend = R_DIM;
      const float* wp = W + (size_t)kbeg * R_DIM + c0;
#pragma unroll 4
      for (int k = kbeg; k < kend; ++k, wp += R_DIM) {
        float  hk = hs[k];
        float4 wv = *reinterpret_cast<const float4*>(wp); // 16B-aligned b128
        ax = fmaf(hk, wv.x, ax);
        ay = fmaf(hk, wv.y, ay);
        az = fmaf(hk, wv.z, az);
        aw = fmaf(hk, wv.w, aw);
      }
    }
    part[tid] = make_float4(ax, ay, az, aw);
    __syncthreads();

    // ---- tree-reduce the 32 k-slices per column group ----
#pragma unroll
    for (int s = 16; s > 0; s >>= 1) {
      if (ks < s) {
        float4 o = part[(ks + s) * 8 + cg];
        float4 m = part[tid];
        m.x += o.x; m.y += o.y; m.z += o.z; m.w += o.w;
        part[tid] = m;
      }
      __syncthreads();
    }

    // ---- writers (wave 0): nonlinearity + leaky update ----
    if (ks == 0 && colvalid) {
      float  xt  = x[t];
      float4 sum = part[cg];
      float  sv[4] = {sum.x, sum.y, sum.z, sum.w};
#pragma unroll
      for (int i = 0; i < 4; ++i) {
        int   col = c0 + i;
        float pre = fmaf(xt, p_w0[i], p_w1[i]) + sv[i];       // u_t + h@W
        float act = tanhf(fmaf(p_g[i], pre, p_b[i]));
        float hnew = fmaf(1.0f - LEAK, hs[col], LEAK * act);
        __builtin_nontemporal_store(hnew, states + (size_t)t * R_DIM + col);
        hbuf[(size_t)((t + 1) & 1) * R_DIM + col] = hnew;     // next-step state
      }
    }

    // ---- device-wide barrier (one per step; h is double-buffered) ----
    if (tid == 0) {
      __threadfence();   // drain wave-0 h stores to L2 (release)
      __hip_atomic_fetch_add(bar, 1u, __ATOMIC_RELAXED, __HIP_MEMORY_SCOPE_AGENT);
      unsigned target = (unsigned)(t + 1) * (unsigned)NB;
      while (__hip_atomic_load(bar, __ATOMIC_RELAXED, __HIP_MEMORY_SCOPE_AGENT) < target)
        __builtin_amdgcn_s_sleep(1);
      __threadfence();   // acquire: invalidate WGP cache before next h read
    }
    __syncthreads();
  }
}

extern "C" void kernel_launch(void* const* d_in, const int* in_sizes, int n_in,
                              void* d_out, int out_size, void* d_ws, size_t ws_size,
                              hipStream_t stream) {
  const float* x    = (const float*)d_in[0];
  const float* Win  = (const float*)d_in[1];
  const float* W    = (const float*)d_in[2];
  const float* gain = (const float*)d_in[3];
  const float* bias = (const float*)d_in[4];
  float* states = (float*)d_out;

  float*    hbuf = (float*)d_ws;                       // 2*R floats = 16000 B
  unsigned* bar  = (unsigned*)((char*)d_ws + 16384);   // barrier counter

  // zero h double-buffer + barrier counter (deterministic per call, capture-safe)
  hipMemsetAsync(d_ws, 0, 16384 + 64, stream);

  esn_persistent<<<NB, BLOCK, 0, stream>>>(x, Win, W, gain, bias, states, hbuf, bar);
}